// VocabParallelDistillLoss_4011499454937
// MI455X (gfx1250) — compile-verified
//
#include <hip/hip_runtime.h>
#include <cstdint>
#include <cstddef>

#define LOG2E 1.4426950408889634f
#define LN2   0.6931471805599453f
#define NBUF  8
#define BLK   256

__device__ __forceinline__ float fast_exp2(float x) {
#if __has_builtin(__builtin_amdgcn_exp2f)
    return __builtin_amdgcn_exp2f(x);
#else
    return exp2f(x);
#endif
}
__device__ __forceinline__ float fast_log2(float x) {
#if __has_builtin(__builtin_amdgcn_logf)
    return __builtin_amdgcn_logf(x);
#else
    return log2f(x);
#endif
}

// gfx1250 async global -> LDS copy, per-lane 16B. Tracked by ASYNCcnt.
__device__ __forceinline__ void async_load_b128(uint32_t lds_off, unsigned long long gaddr) {
    asm volatile("global_load_async_to_lds_b128 %0, %1, off"
                 :: "v"(lds_off), "v"(gaddr)
                 : "memory");
}
__device__ __forceinline__ void wait_async_le7() {
    asm volatile("s_wait_asynccnt 0x7" ::: "memory");
}
__device__ __forceinline__ void wait_async_0() {
    asm volatile("s_wait_asynccnt 0x0" ::: "memory");
}
__device__ __forceinline__ void wait_ds_0() {
    asm volatile("s_wait_dscnt 0x0" ::: "memory");
}

__global__ __launch_bounds__(BLK)
void jsd_topk_kernel(const float* __restrict__ logits,
                     const float* __restrict__ logps,
                     const int*   __restrict__ idx,
                     float*       __restrict__ out,
                     int V, int K)
{
    __shared__ float4 buf[NBUF][BLK];          // 32 KB async streaming buffers
    __shared__ float  red_m[BLK / 32];
    __shared__ float  red_s[BLK / 32];
    __shared__ float  wsum[BLK / 32];
    __shared__ float  s_logZ;

    const int tid = threadIdx.x;
    const int row = blockIdx.x;
    const float* rowp = logits + (size_t)row * (size_t)V;

    const int nvec4 = V >> 2;
    const int nIter = nvec4 / BLK;             // 125 for V=128000

    // Online logsumexp state (base-2 internally; natural log at the end).
    float m  = -3.402823466e+38f;              // running max (avoid inf-inf NaN)
    float s  = 0.f;                            // sum exp(x - m)
    float mL = m * LOG2E;

    const uint32_t lds_base = (uint32_t)(uintptr_t)(&buf[0][tid]);
    const unsigned long long gbase =
        (unsigned long long)(uintptr_t)((const float4*)rowp + tid);

    // -------- async pipeline prologue --------
    const int npro = nIter < NBUF ? nIter : NBUF;
    for (int b = 0; b < npro; ++b) {
        async_load_b128(lds_base + (uint32_t)b * (BLK * 16),
                        gbase + (unsigned long long)b * (BLK * 16));
    }

    // -------- steady-state streaming loop --------
    for (int i = 0; i < nIter; ++i) {
        const int slot = i & (NBUF - 1);
        if (i + NBUF < nIter) wait_async_le7();   // oldest (slot i) complete
        else                  wait_async_0();     // drain tail

        float4 x = buf[slot][tid];                // ds_load_b128

        float xm = fmaxf(fmaxf(x.x, x.y), fmaxf(x.z, x.w));
        if (xm > m) {                              // rare after warm-up
            s *= fast_exp2((m - xm) * LOG2E);
            m  = xm;
            mL = m * LOG2E;
        }
        s += fast_exp2(fmaf(x.x, LOG2E, -mL));
        s += fast_exp2(fmaf(x.y, LOG2E, -mL));
        s += fast_exp2(fmaf(x.z, LOG2E, -mL));
        s += fast_exp2(fmaf(x.w, LOG2E, -mL));

        if (i + NBUF < nIter) {
            wait_ds_0();                           // readback done before overwrite
            async_load_b128(lds_base + (uint32_t)slot * (BLK * 16),
                            gbase + (unsigned long long)(i + NBUF) * (BLK * 16));
        }
    }

    // -------- generic remainder (V not multiple of BLK*4) --------
    for (int j = nIter * BLK * 4 + tid; j < V; j += BLK) {
        float x = rowp[j];
        if (x > m) {
            s *= fast_exp2((m - x) * LOG2E);
            m  = x;
            mL = m * LOG2E;
        }
        s += fast_exp2(fmaf(x, LOG2E, -mL));
    }

    // -------- block reduce (m, s): wave32 shuffle then cross-wave via LDS --------
    for (int off = 16; off > 0; off >>= 1) {
        float mo = __shfl_xor(m, off, 32);
        float so = __shfl_xor(s, off, 32);
        float M  = fmaxf(m, mo);
        s = s * fast_exp2((m - M) * LOG2E) + so * fast_exp2((mo - M) * LOG2E);
        m = M;
    }
    const int wid  = tid >> 5;
    const int lane = tid & 31;
    if (lane == 0) { red_m[wid] = m; red_s[wid] = s; }
    __syncthreads();
    if (tid == 0) {
        float M = red_m[0];
        for (int w = 1; w < BLK / 32; ++w) M = fmaxf(M, red_m[w]);
        float S = 0.f;
        for (int w = 0; w < BLK / 32; ++w) S += red_s[w] * fast_exp2((red_m[w] - M) * LOG2E);
        s_logZ = M + LN2 * fast_log2(S);
    }
    __syncthreads();
    const float logZ = s_logZ;

    // -------- top-K JSD terms (beta = omb = 0.5 folded in) --------
    float c = 0.f;
    if (tid < K) {
        const int   id  = idx  [(size_t)row * K + tid];
        const float lp  = logps[(size_t)row * K + tid];
        const float lq0 = rowp[id] - logZ;                 // ln Q (pre-eps)
        const float Q   = fast_exp2(lq0 * LOG2E);
        const float P   = fast_exp2(lp  * LOG2E);
        const float Mk   = 0.5f * (P + Q);
        const float logM = LN2 * fast_log2(Mk + 1e-20f);
        const float logQ = LN2 * fast_log2(Q  + 1e-20f);   // matches log(Q+EPS)
        c = 0.5f * P * (lp   - logM)
          + 0.5f * Q * (logQ - logM)
          - 0.5f * LN2 * Q;                                // -omb*ln2*Q  (rest term, Q part)
    }
    for (int off = 16; off > 0; off >>= 1) c += __shfl_xor(c, off, 32);
    if (lane == 0) wsum[wid] = c;
    __syncthreads();
    if (tid == 0) {
        float t = 0.f;
        for (int w = 0; w < BLK / 32; ++w) t += wsum[w];
        out[row] = t + 0.5f * LN2;                         // +omb*ln2  (rest term, const part)
    }
}

extern "C" void kernel_launch(void* const* d_in, const int* in_sizes, int n_in,
                              void* d_out, int out_size, void* d_ws, size_t ws_size,
                              hipStream_t stream) {
    (void)n_in; (void)d_ws; (void)ws_size;
    const float* logits = (const float*)d_in[0];
    const float* logps  = (const float*)d_in[1];
    const int*   idx    = (const int*)d_in[2];
    float*       out    = (float*)d_out;

    const int T = out_size;                    // 2048
    const int V = in_sizes[0] / T;             // 128000
    const int K = in_sizes[1] / T;             // 64

    jsd_topk_kernel<<<T, BLK, 0, stream>>>(logits, logps, idx, out, V, K);
}